// STDP_22308060135454
// MI455X (gfx1250) — compile-verified
//
#include <hip/hip_runtime.h>
#include <hip/hip_bf16.h>

// STDP fused kernels for MI455X (gfx1250, wave32, WMMA).
// i = x @ W^T ; spike = (i>=1) ; tp = .5*trace_pre + x ; tq = .5*trace_post + spike
// dw = clip(W,-1,1) * (spike^T @ tp - tq^T @ x)
//
// All GEMMs use v_wmma_f32_16x16x4_f32 (fp32 matrix path) because spike is a
// hard threshold at 1.0 — bf16/f16 inputs would flip spikes near threshold.
// Total HBM traffic ~ W read twice-ish + dw write ~= 35-50 MB -> ~1.5-2 us at
// 23.3 TB/s; 0.8 GFLOP of f32 WMMA is comfortably below that, so the design
// target is pure bandwidth + enough waves to hide latency.

typedef __attribute__((ext_vector_type(2))) float v2f;
typedef __attribute__((ext_vector_type(8))) float v8f;

constexpr int NB   = 32;    // batch
constexpr int PRE  = 2048;
constexpr int POST = 2048;
constexpr int KSPLIT = 16;            // K-chunks for the forward GEMM
constexpr int KCH    = PRE / KSPLIT;  // 128 k-values per chunk

// ---------------------------------------------------------------------------
// tp = 0.5*trace_pre + x   (elementwise, [B, PRE])
// ---------------------------------------------------------------------------
__global__ __launch_bounds__(256) void k_tp(const float* __restrict__ x,
                                            const float* __restrict__ trace_pre,
                                            float* __restrict__ tp_out) {
  int i = blockIdx.x * blockDim.x + threadIdx.x;
  if (i < NB * PRE) tp_out[i] = 0.5f * trace_pre[i] + x[i];
}

// ---------------------------------------------------------------------------
// Forward GEMM, K-split for wave parallelism:
//   partial[chunk] += x[:, kchunk] @ W[:, kchunk]^T   (32 x 16 tile per wave)
// Grid: (POST/128, KSPLIT) blocks x 8 waves = 2048 waves (vs 128 unsplit).
// A (16x4 f32): lanes 0-15 -> M=l with K{k,k+1}; lanes 16-31 -> K{k+2,k+3}.
// B (4x16 f32): lanes 0-15 -> N=l, same K split. Contiguous float2 loads.
// Deterministic: partials land in per-chunk slabs, summed in fixed order.
// ---------------------------------------------------------------------------
__global__ __launch_bounds__(256) void k_gemm_partial(const float* __restrict__ x,
                                                      const float* __restrict__ W,
                                                      float* __restrict__ partial) {
  const int lane  = threadIdx.x & 31;
  const int wave  = threadIdx.x >> 5;                 // 0..7
  const int n0    = (blockIdx.x * 8 + wave) * 16;     // POST tile base
  const int kbase = blockIdx.y * KCH;                 // K chunk base
  const int half  = lane >> 4;
  const int l     = lane & 15;

  v8f c0 = {0.f,0.f,0.f,0.f,0.f,0.f,0.f,0.f};        // batch rows 0..15
  v8f c1 = {0.f,0.f,0.f,0.f,0.f,0.f,0.f,0.f};        // batch rows 16..31

  for (int k0 = 0; k0 < KCH; k0 += 4) {
    const int kk = kbase + k0 + 2 * half;
    v2f a0 = *(const v2f*)(x + (l      ) * PRE + kk);   // A, M=0..15
    v2f a1 = *(const v2f*)(x + (l + 16 ) * PRE + kk);   // A, M=16..31
    v2f bm = *(const v2f*)(W + (n0 + l ) * PRE + kk);   // B = W^T tile
    c0 = __builtin_amdgcn_wmma_f32_16x16x4_f32(false, a0, false, bm, (short)0, c0, false, false);
    c1 = __builtin_amdgcn_wmma_f32_16x16x4_f32(false, a1, false, bm, (short)0, c1, false, false);
  }

  // D layout: lane L holds N = n0 + (L&15); VGPR r holds M = r + 8*(L>=16).
  float* out = partial + blockIdx.y * (NB * POST);
  const int n = n0 + l;
#pragma unroll
  for (int r = 0; r < 8; ++r) {
    const int m0 = r + half * 8;
    out[(m0     ) * POST + n] = c0[r];
    out[(m0 + 16) * POST + n] = c1[r];
  }
}

// ---------------------------------------------------------------------------
// Reduce K-chunks (fixed order -> deterministic), threshold, fuse tq.
// ---------------------------------------------------------------------------
__global__ __launch_bounds__(256) void k_spike_tq(const float* __restrict__ partial,
                                                  const float* __restrict__ trace_post,
                                                  float* __restrict__ spike_out,
                                                  float* __restrict__ tq_out) {
  int i = blockIdx.x * blockDim.x + threadIdx.x;
  if (i < NB * POST) {
    float acc = 0.f;
#pragma unroll
    for (int c = 0; c < KSPLIT; ++c) acc += partial[c * (NB * POST) + i];
    float sp = (acc >= 1.0f) ? 1.0f : 0.0f;
    spike_out[i] = sp;
    tq_out[i]    = 0.5f * trace_post[i] + sp;
  }
}

// ---------------------------------------------------------------------------
// dw tile kernel: one 16x16 (POST x PRE) tile per wave.
//   S1 = tq^T @ x, S2 = spike^T @ tp  (K = batch = 32, 8 steps of 4)
//   dw = clip(W,-1,1) * (S2 - S1)
// All per-lane loads are lane-consecutive (64B coalesced per half-wave).
// 16384 tiles / 8 waves per block -> 2048 blocks: fills the WGPs.
// ---------------------------------------------------------------------------
__global__ __launch_bounds__(256) void k_dw(const float* __restrict__ tq,
                                            const float* __restrict__ x,
                                            const float* __restrict__ spike,
                                            const float* __restrict__ tp,
                                            const float* __restrict__ W,
                                            float* __restrict__ dw) {
  const int lane = threadIdx.x & 31;
  const int wave = threadIdx.x >> 5;
  const int tile = blockIdx.x * 8 + wave;            // 0 .. 16383
  const int qt   = PRE / 16;                         // tiles along PRE
  const int p0   = (tile / qt) * 16;                 // POST base (M)
  const int q0   = (tile % qt) * 16;                 // PRE  base (N)
  const int half = lane >> 4;
  const int l    = lane & 15;

  v8f s1 = {0.f,0.f,0.f,0.f,0.f,0.f,0.f,0.f};
  v8f s2 = {0.f,0.f,0.f,0.f,0.f,0.f,0.f,0.f};

#pragma unroll
  for (int k0 = 0; k0 < NB; k0 += 4) {
    const int kk = k0 + 2 * half;
    v2f a1, a2, b1, b2;
    a1.x = tq   [(kk    ) * POST + p0 + l];
    a1.y = tq   [(kk + 1) * POST + p0 + l];
    a2.x = spike[(kk    ) * POST + p0 + l];
    a2.y = spike[(kk + 1) * POST + p0 + l];
    b1.x = x    [(kk    ) * PRE  + q0 + l];
    b1.y = x    [(kk + 1) * PRE  + q0 + l];
    b2.x = tp   [(kk    ) * PRE  + q0 + l];
    b2.y = tp   [(kk + 1) * PRE  + q0 + l];
    s1 = __builtin_amdgcn_wmma_f32_16x16x4_f32(false, a1, false, b1, (short)0, s1, false, false);
    s2 = __builtin_amdgcn_wmma_f32_16x16x4_f32(false, a2, false, b2, (short)0, s2, false, false);
  }

  // D layout: lane L -> column q0 + (L&15); VGPR r -> row p0 + r + 8*(L>=16).
  const int q = q0 + l;
#pragma unroll
  for (int r = 0; r < 8; ++r) {
    const int p = p0 + r + half * 8;
    float w  = W[p * PRE + q];
    float wc = fminf(fmaxf(w, -1.0f), 1.0f);
    dw[p * PRE + q] = wc * (s2[r] - s1[r]);
  }
}

extern "C" void kernel_launch(void* const* d_in, const int* in_sizes, int n_in,
                              void* d_out, int out_size, void* d_ws, size_t ws_size,
                              hipStream_t stream) {
  (void)in_sizes; (void)n_in; (void)out_size; (void)ws_size;
  const float* x          = (const float*)d_in[0];
  const float* W          = (const float*)d_in[1];
  const float* trace_pre  = (const float*)d_in[2];
  const float* trace_post = (const float*)d_in[3];

  float* spike_out = (float*)d_out;                 // [B, POST]
  float* dw_out    = spike_out + NB * POST;         // [POST, PRE]

  float* tp      = (float*)d_ws;                    // [B, PRE]
  float* tq      = tp + NB * PRE;                   // [B, POST]
  float* partial = tq + NB * POST;                  // [KSPLIT, B, POST]

  // tp has no dependency on the GEMM; launch first.
  k_tp<<<(NB * PRE + 255) / 256, 256, 0, stream>>>(x, trace_pre, tp);

  // Forward GEMM, K-split: (16, 16) blocks x 8 waves = 2048 waves.
  {
    dim3 grid(POST / (16 * 8), KSPLIT);
    k_gemm_partial<<<grid, 256, 0, stream>>>(x, W, partial);
  }

  // Reduce partials -> spike (d_out) + tq (ws).
  k_spike_tq<<<(NB * POST + 255) / 256, 256, 0, stream>>>(partial, trace_post, spike_out, tq);

  // dw: 128x128 = 16384 16x16 tiles, 8 waves per block -> 2048 blocks.
  k_dw<<<(POST / 16) * (PRE / 16) / 8, 256, 0, stream>>>(tq, x, spike_out, tp, W, dw_out);
}